// PathGuidedAggregator_1202590843113
// MI455X (gfx1250) — compile-verified
//
#include <hip/hip_runtime.h>

#define DIM   128
#define CHUNK 512
#define BLOCK 256
#define WAVES (BLOCK / 32)
#define EPW   (CHUNK / WAVES) // 64 edges per wave
#define PF    16              // prefetch distance (edges ahead)

// On gfx1250 the low 32 bits of a flat shared-memory pointer are the LDS byte
// offset (ISA 10.2); async-to-LDS VDST wants that.
__device__ __forceinline__ unsigned lds_off(const void* p) {
    return (unsigned)(unsigned long long)p;
}

__global__ void __launch_bounds__(BLOCK)
pga_agg_kernel(const float* __restrict__ embeds,
               const int* __restrict__ rows,
               const int* __restrict__ cols,
               float* __restrict__ out,
               unsigned* __restrict__ cnt,
               int nEdges) {
    __shared__ int sh_rows[CHUNK];
    __shared__ int sh_cols[CHUNK];

    const int tid = threadIdx.x;
    const long long base = (long long)blockIdx.x * CHUNK;
    long long rem = (long long)nEdges - base;
    const int cend = (rem < CHUNK) ? (rem < 0 ? 0 : (int)rem) : CHUNK;

    // Stream edge indices into LDS via CDNA5 async global->LDS loads
    // (ASYNCcnt path; no VGPR round-trip for the index data).
    for (int k = tid; k < cend; k += BLOCK) {
        unsigned lr = lds_off(&sh_rows[k]);
        unsigned lc = lds_off(&sh_cols[k]);
        unsigned go = (unsigned)((base + (long long)k) * 4ll); // byte offset
        asm volatile("global_load_async_to_lds_b32 %0, %1, %2"
                     :: "v"(lr), "v"(go), "s"(rows) : "memory");
        asm volatile("global_load_async_to_lds_b32 %0, %1, %2"
                     :: "v"(lc), "v"(go), "s"(cols) : "memory");
    }
    asm volatile("s_wait_asynccnt 0" ::: "memory");
    __syncthreads();

    // wave32 layout: one wave owns all 128 dims (float4 per lane), walks a
    // contiguous 64-edge run of the sorted row list. Row value is uniform
    // across the wave -> flush branch is non-divergent.
    const int wave = tid >> 5;
    const int lane = tid & 31;
    int e = wave * EPW;
    int eend = e + EPW;
    if (eend > cend) eend = cend;
    if (e >= eend) return;

    const int d0 = lane * 4;

    // Prime the prefetch window: pull upcoming gather rows toward the WGP
    // (global_prefetch_b8; fetches the whole cacheline, no counters).
    for (int p = e; p < e + PF && p < eend; ++p) {
        const float* pp = embeds + (long long)sh_cols[p] * DIM + d0;
        __builtin_prefetch(pp, 0, 3);
    }

    float4 acc = make_float4(0.f, 0.f, 0.f, 0.f);
    int cur = sh_rows[e];
    int run = 0;

    for (; e < eend; ++e) {
        // keep the prefetch pipeline PF edges ahead of the dependent load
        {
            int p = e + PF;
            if (p >= eend) p = eend - 1;
            const float* pp = embeds + (long long)sh_cols[p] * DIM + d0;
            __builtin_prefetch(pp, 0, 3);
        }
        const int r = sh_rows[e];   // LDS broadcast read
        const int c = sh_cols[e];   // LDS broadcast read
        if (r != cur) {
            float* dst = out + (long long)cur * DIM + d0;
            atomicAdd(dst + 0, acc.x);
            atomicAdd(dst + 1, acc.y);
            atomicAdd(dst + 2, acc.z);
            atomicAdd(dst + 3, acc.w);
            if (lane == 0) atomicAdd(&cnt[cur], (unsigned)run);
            acc = make_float4(0.f, 0.f, 0.f, 0.f);
            run = 0;
            cur = r;
        }
        // one fully-coalesced 512B wave load per edge (L2-resident table)
        const float4 v =
            *reinterpret_cast<const float4*>(embeds + (long long)c * DIM + d0);
        acc.x += v.x; acc.y += v.y; acc.z += v.z; acc.w += v.w;
        ++run;
    }
    float* dst = out + (long long)cur * DIM + d0;
    atomicAdd(dst + 0, acc.x);
    atomicAdd(dst + 1, acc.y);
    atomicAdd(dst + 2, acc.z);
    atomicAdd(dst + 3, acc.w);
    if (lane == 0) atomicAdd(&cnt[cur], (unsigned)run);
}

__global__ void __launch_bounds__(BLOCK)
pga_zero_kernel(float4* __restrict__ out4, unsigned* __restrict__ cnt,
                int n4, int nc) {
    const int stride = gridDim.x * blockDim.x;
    for (int i = blockIdx.x * blockDim.x + threadIdx.x; i < n4; i += stride)
        out4[i] = make_float4(0.f, 0.f, 0.f, 0.f);
    for (int i = blockIdx.x * blockDim.x + threadIdx.x; i < nc; i += stride)
        cnt[i] = 0u;
}

__global__ void __launch_bounds__(BLOCK)
pga_finalize_kernel(float* __restrict__ out,
                    const unsigned* __restrict__ cnt,
                    const int* __restrict__ deg,
                    int total) {
    const int stride = gridDim.x * blockDim.x;
    for (int i = blockIdx.x * blockDim.x + threadIdx.x; i < total; i += stride) {
        const int row = i >> 7; // / DIM
        const unsigned c = cnt[row];
        const float scale =
            (deg[row] <= 5) ? (1.0f / (float)(c > 1u ? c : 1u)) : 0.0f;
        out[i] *= scale;
    }
}

extern "C" void kernel_launch(void* const* d_in, const int* in_sizes, int n_in,
                              void* d_out, int out_size, void* d_ws, size_t ws_size,
                              hipStream_t stream) {
    const float* embeds = (const float*)d_in[0];   // [N, 128] f32
    const int*   deg    = (const int*)d_in[1];     // [N]
    const int*   rows   = (const int*)d_in[2];     // [E] sorted
    const int*   cols   = (const int*)d_in[3];     // [E]
    float* out = (float*)d_out;                    // [N, 128] f32
    unsigned* cnt = (unsigned*)d_ws;               // [N] edge counts

    const int N = in_sizes[1];
    const int E = in_sizes[2];
    const int total = N * DIM;

    pga_zero_kernel<<<1024, BLOCK, 0, stream>>>((float4*)out, cnt, total / 4, N);

    const int nblk = (E + CHUNK - 1) / CHUNK;
    pga_agg_kernel<<<nblk, BLOCK, 0, stream>>>(embeds, rows, cols, out, cnt, E);

    pga_finalize_kernel<<<1024, BLOCK, 0, stream>>>(out, cnt, deg, total);
}